// KNN_43267500540661
// MI455X (gfx1250) — compile-verified
//
#include <hip/hip_runtime.h>

#define KNN_K 16

typedef __attribute__((ext_vector_type(2))) float v2f;
typedef __attribute__((ext_vector_type(8))) float v8f;

// Pack barycenters [N,3] -> float4 (x, y, z, |p|^2): b128 loads + norm folded
// into the WMMA K=4 slot (score = |c|^2 - 2 q.c == dist^2 - |q|^2, same order).
__global__ void knn_pack_kernel(const float* __restrict__ bary,
                                float4* __restrict__ P, int n) {
  int i = blockIdx.x * blockDim.x + threadIdx.x;
  if (i < n) {
    float x = bary[3 * i + 0];
    float y = bary[3 * i + 1];
    float z = bary[3 * i + 2];
    P[i] = make_float4(x, y, z, x * x + y * y + z * z);
  }
}

// Branch-free sorted insertion: compare-exchange chain of v_cmp + v_cndmask.
// No exec-branches inside -> no whole-array mov storms. Outer guard branch
// skips the (common) no-insert case. Strict '<' keeps earliest index on ties.
__device__ __forceinline__ void knn_insert(float s, int id,
                                           float* dk, int* ik) {
  if (s < dk[KNN_K - 1]) {
    float ds = s;
    int di = id;
#pragma unroll
    for (int j = 0; j < KNN_K; ++j) {
      bool c = ds < dk[j];
      float td = dk[j];
      int ti = ik[j];
      dk[j] = c ? ds : td;
      ik[j] = c ? di : ti;
      ds = c ? td : ds;
      di = c ? ti : di;
    }
  }
}

// One wave32 per 16-query tile; loop over candidate tiles 2 at a time:
// 2x global_load_b128 + 2x V_WMMA_F32_16X16X4_F32 -> two 16x16 score tiles,
// min-tree reject, register-resident sorted top-16 per lane (disjoint halves).
__global__ void __launch_bounds__(256)
knn_main_kernel(const float4* __restrict__ P, float* __restrict__ out, int n) {
  const int lane = threadIdx.x & 31;
  const int wave = threadIdx.x >> 5;
  const int qtile = blockIdx.x * (blockDim.x >> 5) + wave;
  const int numTiles = (n + 15) >> 4;
  if (qtile >= numTiles) return;  // wave-uniform: EXEC all-1s at every WMMA

  const int qbase = qtile * 16;
  const int sub = lane & 15;
  const bool hi = lane >= 16;
  const float INF = __builtin_inff();

  // B operand (queries, 4x16): lanes 0-15 (K0,K1)=(-2qx,-2qy),
  //                            lanes 16-31 (K2,K3)=(-2qz, 1)
  float4 qp = P[min(qbase + sub, n - 1)];
  v2f b;
  b.x = hi ? (-2.0f * qp.z) : (-2.0f * qp.x);
  b.y = hi ? 1.0f : (-2.0f * qp.y);

  float dk[KNN_K];
  int ik[KNN_K];
#pragma unroll
  for (int j = 0; j < KNN_K; ++j) { dk[j] = INF; ik[j] = 0; }

  const int mOff = hi ? 8 : 0;  // C/D rows: lanes<16 -> M=0..7, else M=8..15

  // A operand (candidates, 16x4): lanes 0-15 (K0,K1)=(x,y),
  // lanes 16-31 (K2,K3)=(z,|c|^2). Rows with cm>=n get |c|^2=+INF so their
  // score is +INF and never inserts -> tail tiles need no guard, EXEC uniform.
  auto makeA = [&](int cm, float4 cp) -> v2f {
    v2f a;
    a.x = hi ? cp.z : cp.x;
    a.y = hi ? ((cm < n) ? cp.w : INF) : cp.y;
    return a;
  };

  for (int ct = 0; ct < numTiles; ct += 2) {
    const int cm0 = ct * 16 + sub;
    const int cm1 = cm0 + 16;
    float4 cp0 = P[min(cm0, n - 1)];
    float4 cp1 = P[min(cm1, n - 1)];
    if (ct + 2 < numTiles)  // one prefetch covers both next tiles (32 float4)
      __builtin_prefetch(&P[min(ct * 16 + 32 + lane, n - 1)], 0, 3);

    v2f a0 = makeA(cm0, cp0);
    v2f a1 = makeA(cm1, cp1);
    v8f z = {};
    v8f s0 = __builtin_amdgcn_wmma_f32_16x16x4_f32(false, a0, false, b,
                                                   (short)0, z, false, false);
    v8f s1 = __builtin_amdgcn_wmma_f32_16x16x4_f32(false, a1, false, b,
                                                   (short)0, z, false, false);

    // Reject both tiles with one compare in the common case.
    float t0 = fminf(fminf(fminf(s0[0], s0[1]), fminf(s0[2], s0[3])),
                     fminf(fminf(s0[4], s0[5]), fminf(s0[6], s0[7])));
    float t1 = fminf(fminf(fminf(s1[0], s1[1]), fminf(s1[2], s1[3])),
                     fminf(fminf(s1[4], s1[5]), fminf(s1[6], s1[7])));
    if (fminf(t0, t1) < dk[KNN_K - 1]) {
      const int mb0 = ct * 16 + mOff;
#pragma unroll
      for (int v = 0; v < 8; ++v) knn_insert(s0[v], mb0 + v, dk, ik);
#pragma unroll
      for (int v = 0; v < 8; ++v) knn_insert(s1[v], mb0 + 16 + v, dk, ik);
    }
  }

  // Merge lane l (M=0..7 halves) with lane l+16 (M=8..15 halves): partner's
  // sorted list arrives via 16 shuffles, absorbed by guarded insertions.
#pragma unroll
  for (int j = 0; j < KNN_K; ++j) {
    float od = __shfl(dk[j], sub + 16);
    int oi = __shfl(ik[j], sub + 16);
    if (!hi) knn_insert(od, oi, dk, ik);
  }

  if (!hi) {
    int q = qbase + sub;
    if (q < n) {
#pragma unroll
      for (int j = 0; j < KNN_K; ++j) out[q * KNN_K + j] = (float)ik[j];
    }
  }
}

extern "C" void kernel_launch(void* const* d_in, const int* in_sizes, int n_in,
                              void* d_out, int out_size, void* d_ws,
                              size_t ws_size, hipStream_t stream) {
  (void)n_in; (void)out_size; (void)ws_size;
  const float* bary = (const float*)d_in[0];
  const int n = in_sizes[0] / 3;

  float4* P = (float4*)d_ws;  // n * 16 bytes of scratch
  knn_pack_kernel<<<(n + 255) / 256, 256, 0, stream>>>(bary, P, n);

  const int qtiles = (n + 15) / 16;
  const int wavesPerBlock = 256 / 32;
  const int blocks = (qtiles + wavesPerBlock - 1) / wavesPerBlock;
  knn_main_kernel<<<blocks, 256, 0, stream>>>(P, (float*)d_out, n);
}